// UNIFIED_DilatedSpatioTemporalGCN_60129542621
// MI455X (gfx1250) — compile-verified
//
#include <hip/hip_runtime.h>
#include <math.h>

// Problem constants (from reference)
#define B_  16
#define T_  12
#define F_  32
#define N_  500
#define L_  3
#define NTILES 32     // ceil(500/16) -> padded to 512 columns
#define SA  34        // LDS stride for 32x32 A matrices (bank-conflict free v2f reads)
#define ST  18        // LDS stride for per-wave 32x16 staging tile

typedef float v2f __attribute__((ext_vector_type(2)));
typedef float v8f __attribute__((ext_vector_type(8)));

// D(16x16,f32) += A(16x4,f32) * B(4x16,f32); wave32 striped operands.
__device__ __forceinline__ v8f wmma_f32_k4(v2f a, v2f b, v8f c) {
  return __builtin_amdgcn_wmma_f32_16x16x4_f32(false, a, false, b, (short)0, c,
                                               false, false);
}

__device__ __forceinline__ v8f vzero8() {
  v8f z = {0.f,0.f,0.f,0.f,0.f,0.f,0.f,0.f};
  return z;
}

// ---------------------------------------------------------------------------
// Kernel 1: per (b,t) slab. propX^T = (colsum + X)/(N+1) folded into B-frag
// load; gcn1^T = gcn_w^T @ propX^T + b1 (staged in per-wave LDS tile);
// gcn2^T = sigmoid(gate_w^T @ gcn1^T + b2) -> global [b,t,f,n].
// ---------------------------------------------------------------------------
__global__ __launch_bounds__(256) void k_gcn(
    const float* __restrict__ x,      // [B,T,F,N]
    const float* __restrict__ gw,     // [F,F] this layer
    const float* __restrict__ gb,     // [F]  this layer
    const float* __restrict__ gatew,  // [F,F]
    const float* __restrict__ gateb,  // [F]
    float* __restrict__ gout)         // [B,T,F,N]
{
  __shared__ float sA1[F_*SA];        // gcn_w^T  : sA1[row=g][k=f]
  __shared__ float sA2[F_*SA];        // gate_w^T
  __shared__ float sb1[F_], sb2[F_], ssum[F_];
  __shared__ float stile[8*F_*ST];    // per-wave 32x16 gcn1 staging

  const int bt = blockIdx.x;
  const float* xp = x + (size_t)bt*F_*N_;
  float* op = gout + (size_t)bt*F_*N_;
  const int tid = threadIdx.x;

  for (int i = tid; i < F_*F_; i += 256) {
    int f = i >> 5, g = i & 31;
    sA1[g*SA + f] = gw[f*F_ + g];
    sA2[g*SA + f] = gatew[f*F_ + g];
  }
  if (tid < F_) { sb1[tid] = gb[tid]; sb2[tid] = gateb[tid]; ssum[tid] = 0.0f; }
  __syncthreads();
  {  // column sums over N for each feature f
    int f = tid >> 3, p = tid & 7;
    float s = 0.0f;
    for (int n = p; n < N_; n += 8) s += xp[f*N_ + n];
    atomicAdd(&ssum[f], s);          // ds_add_f32
  }
  __syncthreads();

  const float inv = 1.0f / (float)(N_ + 1);
  const int wave = tid >> 5, lane = tid & 31;
  const int hi = lane >> 4, col = lane & 15;
  float* mytile = &stile[wave*F_*ST];

  for (int ct = wave; ct < NTILES; ct += 8) {      // uniform loop: EXEC full
    const int n = ct*16 + col;
    const int nidx = (n < N_) ? n : (N_-1);        // clamp; columns independent
    v8f acc0 = vzero8(), acc1 = vzero8();
#pragma unroll
    for (int s = 0; s < 8; ++s) {
      const int k = 4*s + 2*hi;
      v2f bfr;
      bfr.x = (xp[(k  )*N_ + nidx] + ssum[k  ]) * inv;
      bfr.y = (xp[(k+1)*N_ + nidx] + ssum[k+1]) * inv;
      v2f a0 = *(const v2f*)&sA1[(     col)*SA + k];
      acc0 = wmma_f32_k4(a0, bfr, acc0);
      v2f a1 = *(const v2f*)&sA1[(16 + col)*SA + k];
      acc1 = wmma_f32_k4(a1, bfr, acc1);
    }
#pragma unroll
    for (int vr = 0; vr < 8; ++vr) {               // stage gcn1 (+bias) in LDS
      const int m = vr + hi*8;
      mytile[(     m)*ST + col] = acc0[vr] + sb1[m];
      mytile[(16 + m)*ST + col] = acc1[vr] + sb1[16 + m];
    }
    acc0 = vzero8(); acc1 = vzero8();
#pragma unroll
    for (int s = 0; s < 8; ++s) {                  // gate GEMM, B from LDS
      const int k = 4*s + 2*hi;
      v2f bfr;
      bfr.x = mytile[(k  )*ST + col];
      bfr.y = mytile[(k+1)*ST + col];
      v2f a0 = *(const v2f*)&sA2[(     col)*SA + k];
      acc0 = wmma_f32_k4(a0, bfr, acc0);
      v2f a1 = *(const v2f*)&sA2[(16 + col)*SA + k];
      acc1 = wmma_f32_k4(a1, bfr, acc1);
    }
    if (n < N_) {
#pragma unroll
      for (int vr = 0; vr < 8; ++vr) {
        const int m = vr + hi*8;
        float v0 = acc0[vr] + sb2[m];
        op[(     m)*N_ + n] = 1.0f/(1.0f + __expf(-v0));
        float v1 = acc1[vr] + sb2[16 + m];
        op[(16 + m)*N_ + n] = 1.0f/(1.0f + __expf(-v1));
      }
    }
  }
}

// ---------------------------------------------------------------------------
// Kernel 2: dilated temporal conv (K=2) + bias + ReLU + residual; captures
// res_agg at t = T-1.  out^T = w(k=1)^T@gcn2(t) + w(k=0)^T@gcn2(t-dil).
// ---------------------------------------------------------------------------
__global__ __launch_bounds__(256) void k_conv(
    const float* __restrict__ gcn2,   // [B,T,F,N]
    const float* __restrict__ xin,    // [B,T,F,N] residual
    const float* __restrict__ cw,     // conv_w[li]: [F][F][1][2]
    const float* __restrict__ cb,     // conv_b[li]: [F]
    float* __restrict__ xout,         // [B,T,F,N]
    float* __restrict__ resl,         // [B,F,N] slice for this layer
    const int dil, const int write_x)
{
  __shared__ float sW1[F_*SA];        // k=1 tap (current t)
  __shared__ float sW0[F_*SA];        // k=0 tap (t - dil)
  __shared__ float scb[F_];
  const int bt = blockIdx.x;
  const int t = bt % T_, b = bt / T_;
  const int tid = threadIdx.x;
  for (int i = tid; i < F_*F_; i += 256) {
    int o = i >> 5, ii = i & 31;
    sW1[o*SA + ii] = cw[(o*F_ + ii)*2 + 1];
    sW0[o*SA + ii] = cw[(o*F_ + ii)*2 + 0];
  }
  if (tid < F_) scb[tid] = cb[tid];
  __syncthreads();

  const float* g1 = gcn2 + (size_t)bt*F_*N_;
  const float* g0 = gcn2 + (size_t)(bt - dil)*F_*N_;   // used only if t>=dil
  const float* xr = xin  + (size_t)bt*F_*N_;
  float* xo = xout + (size_t)bt*F_*N_;

  const int wave = tid >> 5, lane = tid & 31;
  const int hi = lane >> 4, col = lane & 15;

  for (int ct = wave; ct < NTILES; ct += 8) {
    const int n = ct*16 + col;
    const int nidx = (n < N_) ? n : (N_-1);
    v8f acc0 = vzero8(), acc1 = vzero8();
#pragma unroll
    for (int s = 0; s < 8; ++s) {
      const int k = 4*s + 2*hi;
      v2f bfr;
      bfr.x = g1[(k  )*N_ + nidx];
      bfr.y = g1[(k+1)*N_ + nidx];
      v2f a0 = *(const v2f*)&sW1[(     col)*SA + k];
      acc0 = wmma_f32_k4(a0, bfr, acc0);
      v2f a1 = *(const v2f*)&sW1[(16 + col)*SA + k];
      acc1 = wmma_f32_k4(a1, bfr, acc1);
    }
    if (t >= dil) {                    // block-uniform branch: EXEC stays full
#pragma unroll
      for (int s = 0; s < 8; ++s) {
        const int k = 4*s + 2*hi;
        v2f bfr;
        bfr.x = g0[(k  )*N_ + nidx];
        bfr.y = g0[(k+1)*N_ + nidx];
        v2f a0 = *(const v2f*)&sW0[(     col)*SA + k];
        acc0 = wmma_f32_k4(a0, bfr, acc0);
        v2f a1 = *(const v2f*)&sW0[(16 + col)*SA + k];
        acc1 = wmma_f32_k4(a1, bfr, acc1);
      }
    }
    if (n < N_) {
#pragma unroll
      for (int vr = 0; vr < 8; ++vr) {
        const int m = vr + hi*8;
        float o0 = fmaxf(acc0[vr] + scb[m],      0.0f);
        float o1 = fmaxf(acc1[vr] + scb[16 + m], 0.0f);
        if (write_x) {
          xo[(     m)*N_ + n] = o0 + xr[(     m)*N_ + n];
          xo[(16 + m)*N_ + n] = o1 + xr[(16 + m)*N_ + n];
        }
        if (t == T_-1) {
          resl[(size_t)(b*F_ +      m)*N_ + n] = o0;
          resl[(size_t)(b*F_ + 16 + m)*N_ + n] = o1;
        }
      }
    }
  }
}

// ---------------------------------------------------------------------------
// Kernel 3: attention scores per (b,l): h^T = tanh(Wa^T @ res^T + ba);
// s[b, l*N+n] = sum_f h^T[f][n]*v[f]  (flat (b,l,n) order, as the reference's
// pre-reshape buffer).
// ---------------------------------------------------------------------------
__global__ __launch_bounds__(256) void k_att(
    const float* __restrict__ resl,   // [L,B,F,N]
    const float* __restrict__ Wa,     // [F,F]
    const float* __restrict__ ba,     // [F]
    const float* __restrict__ vvec,   // [F]
    float* __restrict__ sbuf)         // [B, L*N]
{
  __shared__ float sA[F_*SA];         // Wa^T : sA[row=f][k=g]
  __shared__ float sba[F_], sv[F_];
  __shared__ float ssv[NTILES*16];
  const int bl = blockIdx.x;
  const int l = bl % L_, b = bl / L_;
  const int tid = threadIdx.x;
  for (int i = tid; i < F_*F_; i += 256) {
    int f = i >> 5, g = i & 31;
    sA[f*SA + g] = Wa[g*F_ + f];
  }
  if (tid < F_) { sba[tid] = ba[tid]; sv[tid] = vvec[tid]; }
  for (int i = tid; i < NTILES*16; i += 256) ssv[i] = 0.0f;
  __syncthreads();

  const float* bp = resl + (size_t)(l*B_ + b)*F_*N_;
  const int wave = tid >> 5, lane = tid & 31;
  const int hi = lane >> 4, col = lane & 15;

  for (int ct = wave; ct < NTILES; ct += 8) {
    const int n = ct*16 + col;
    const int nidx = (n < N_) ? n : (N_-1);
    v8f acc0 = vzero8(), acc1 = vzero8();
#pragma unroll
    for (int s = 0; s < 8; ++s) {
      const int k = 4*s + 2*hi;
      v2f bfr;
      bfr.x = bp[(k  )*N_ + nidx];
      bfr.y = bp[(k+1)*N_ + nidx];
      v2f a0 = *(const v2f*)&sA[(     col)*SA + k];
      acc0 = wmma_f32_k4(a0, bfr, acc0);
      v2f a1 = *(const v2f*)&sA[(16 + col)*SA + k];
      acc1 = wmma_f32_k4(a1, bfr, acc1);
    }
    float part = 0.0f;                 // this lane covers 16 of the 32 f-rows
#pragma unroll
    for (int vr = 0; vr < 8; ++vr) {
      const int m = vr + hi*8;
      part += tanhf(acc0[vr] + sba[m])      * sv[m];
      part += tanhf(acc1[vr] + sba[16 + m]) * sv[16 + m];
    }
    atomicAdd(&ssv[ct*16 + col], part);
  }
  __syncthreads();
  for (int n = tid; n < N_; n += 256)
    sbuf[(size_t)b*(L_*N_) + l*N_ + n] = ssv[n];
}

// ---------------------------------------------------------------------------
// Kernel 4: reference's flat reshape (B,L,N)->(B,N,L), 3-way softmax, combine.
// ---------------------------------------------------------------------------
__global__ __launch_bounds__(256) void k_final(
    const float* __restrict__ sbuf,   // [B, L*N]
    const float* __restrict__ resl,   // [L,B,F,N]
    float* __restrict__ y)            // [B,N,F]
{
  const int idx = blockIdx.x*256 + threadIdx.x;
  if (idx >= B_*N_*F_) return;
  const int f = idx & (F_-1);
  const int n = (idx >> 5) % N_;
  const int b = idx / (N_*F_);
  const size_t sb = (size_t)b*(L_*N_) + (size_t)n*L_;   // reshape, not transpose
  float s0 = sbuf[sb + 0], s1 = sbuf[sb + 1], s2 = sbuf[sb + 2];
  float mx = fmaxf(s0, fmaxf(s1, s2));
  float e0 = __expf(s0 - mx), e1 = __expf(s1 - mx), e2 = __expf(s2 - mx);
  float inv = 1.0f/(e0 + e1 + e2);
  float r0 = resl[((size_t)(0*B_ + b)*F_ + f)*N_ + n];
  float r1 = resl[((size_t)(1*B_ + b)*F_ + f)*N_ + n];
  float r2 = resl[((size_t)(2*B_ + b)*F_ + f)*N_ + n];
  y[idx] = (e0*r0 + e1*r1 + e2*r2) * inv;
}

extern "C" void kernel_launch(void* const* d_in, const int* in_sizes, int n_in,
                              void* d_out, int out_size, void* d_ws, size_t ws_size,
                              hipStream_t stream) {
  (void)in_sizes; (void)n_in; (void)out_size; (void)ws_size;
  // Inputs in setup_inputs() order; B_mat/static_MTE/learnable_adj/flags are
  // dead code in the reference and intentionally unused.
  const float* x_in   = (const float*)d_in[0];
  const float* conv_w = (const float*)d_in[4];   // [L,F,F,1,2]
  const float* conv_b = (const float*)d_in[5];   // [L,F]
  const float* gcn_w  = (const float*)d_in[6];   // [L,F,F]
  const float* gcn_b  = (const float*)d_in[7];   // [L,F]
  const float* gate_w = (const float*)d_in[8];   // [F,F]
  const float* gate_b = (const float*)d_in[9];   // [F]
  const float* att_Wa = (const float*)d_in[10];  // [F,F]
  const float* att_ba = (const float*)d_in[11];  // [F]
  const float* att_v  = (const float*)d_in[12];  // [F,1]

  float* ws = (float*)d_ws;
  const size_t SLAB = (size_t)B_*T_*F_*N_;       // 3,072,000 floats
  float* gcnbuf = ws;                            // gated gcn output, reused per layer
  float* xb0    = ws + SLAB;                     // layer ping
  float* xb1    = ws + 2*SLAB;                   // layer pong
  float* resl   = ws + 3*SLAB;                   // [L,B,F,N] = 768,000
  float* sbuf   = resl + (size_t)L_*B_*F_*N_;    // [B,L*N]   =  24,000

  const int DILS[L_] = {1, 2, 4};
  dim3 blk(256);
  dim3 gridBT(B_*T_);

  const float* xcur = x_in;
  float* xnext[L_] = {xb0, xb1, xb1 /*unused (write_x=0)*/};
  for (int li = 0; li < L_; ++li) {
    k_gcn<<<gridBT, blk, 0, stream>>>(xcur, gcn_w + li*F_*F_, gcn_b + li*F_,
                                      gate_w, gate_b, gcnbuf);
    k_conv<<<gridBT, blk, 0, stream>>>(gcnbuf, xcur,
                                       conv_w + li*F_*F_*2, conv_b + li*F_,
                                       xnext[li], resl + (size_t)li*B_*F_*N_,
                                       DILS[li], (li < L_-1) ? 1 : 0);
    xcur = xnext[li];
  }
  k_att<<<dim3(B_*L_), blk, 0, stream>>>(resl, att_Wa, att_ba, att_v, sbuf);
  k_final<<<dim3((B_*N_*F_ + 255)/256), blk, 0, stream>>>(sbuf, resl, (float*)d_out);
}